// TropicalMinReLu_70832600646267
// MI455X (gfx1250) — compile-verified
//
#include <hip/hip_runtime.h>
#include <hip/hip_bf16.h>
#include <math.h>

// Tropical (min,+) "GEMM": out[b,o] = min_i ( relu_inf(X[b,i]) + W[o,i] )
// B=1024, O=512, I=512, fp32. VALU-bound (WMMA is a (+,x) engine; min-plus
// semiring cannot use it). 64x64 output tile / 256 threads, split-K=2 for
// occupancy (256 blocks), K=32 chunks staged via ASYNC global->LDS copies
// (double-buffered, overlapped with compute), XOR-swizzled k-major tiles for
// conflict-free ds_load_b128, 4x4 register tile per thread. Inner loop is
// one v_add_f32 + one v_min_num_f32 per MAC: measured to VOPD-pair best
// (round 3: 1305 valu instrs vs 1624 for the min3 variant). K halves are
// combined with fp-min atomics (exact, order-independent).

#define TILE   64
#define KCH    32
#define SPLITK 2
#define INF_F (__builtin_inff())

constexpr int IN_F = 512;   // reference IN_FEATURES (K dimension)

#ifndef __has_builtin
#define __has_builtin(x) 0
#endif

#if __has_builtin(__builtin_amdgcn_global_load_async_to_lds_b128) && \
    __has_builtin(__builtin_amdgcn_s_wait_asynccnt)
#define USE_ASYNC_LDS 1
#else
#define USE_ASYNC_LDS 0
#endif

typedef int v4i_t __attribute__((ext_vector_type(4)));
typedef __attribute__((address_space(3))) v4i_t* lds_v4i_ptr;

__device__ __forceinline__ void atomic_fmin(float* p, float v) {
#if __has_builtin(__hip_atomic_fetch_min)
  __hip_atomic_fetch_min(p, v, __ATOMIC_RELAXED, __HIP_MEMORY_SCOPE_AGENT);
#else
  unsigned int* ip = (unsigned int*)p;
  unsigned int old = *ip;
  for (;;) {
    float f = __uint_as_float(old);
    if (f <= v) break;
    unsigned int assumed = old;
    old = atomicCAS(ip, assumed, __float_as_uint(v));
    if (old == assumed) break;
  }
#endif
}

__global__ __launch_bounds__(256)
void init_inf_kernel(float* __restrict__ out, int n4) {
  int i = blockIdx.x * blockDim.x + threadIdx.x;
  if (i < n4) {
    float4 v;
    v.x = INF_F; v.y = INF_F; v.z = INF_F; v.w = INF_F;
    *(float4*)(out + (size_t)i * 4) = v;
  }
}

__global__ __launch_bounds__(256)
void TropicalMinReLu_kernel(const float* __restrict__ X,
                            const float* __restrict__ W,
                            float* __restrict__ out,
                            int B, int O) {
  // Compute tiles, k-major + XOR-8 swizzle: (row,k) -> Ts[k][row ^ ((k>>2)<<3)]
  __shared__ float Xs[KCH][TILE];          // 8 KB
  __shared__ float Ws[KCH][TILE];          // 8 KB
#if USE_ASYNC_LDS
  __shared__ float4 XrStage[2][512];       // 16 KB double-buffered raw stage
  __shared__ float4 WrStage[2][512];       // 16 KB
#endif

  const int t   = threadIdx.x;                    // 0..255
  const int b0  = blockIdx.y * TILE;              // batch-tile origin
  const int o0  = blockIdx.x * TILE;              // out-feature-tile origin
  const int kz0 = blockIdx.z * (IN_F / SPLITK);   // this block's K origin
  const int nch = (IN_F / SPLITK) / KCH;          // 8 chunks per K half

  // loader mapping: 512 float4 slots/tile, 2 per thread.
  // idx: row = idx>>3 (0..63), kq = idx&7 -> 8 lanes stream 128B of one row.
  int lrow[2], lkq[2];
  const float* xptr[2];
  const float* wptr[2];
#pragma unroll
  for (int i = 0; i < 2; ++i) {
    int idx = t + 256 * i;
    lrow[i] = idx >> 3;
    lkq[i]  = idx & 7;
    xptr[i] = X + (size_t)(b0 + lrow[i]) * IN_F + kz0 + lkq[i] * 4;
    wptr[i] = W + (size_t)(o0 + lrow[i]) * IN_F + kz0 + lkq[i] * 4;
  }

  const int tx = t & 15;
  const int ty = t >> 4;
  float acc[4][4];
#pragma unroll
  for (int i = 0; i < 4; ++i)
#pragma unroll
    for (int j = 0; j < 4; ++j)
      acc[i][j] = INF_F;

  // ---- helpers -------------------------------------------------------------
#if USE_ASYNC_LDS
  auto issue_chunk = [&](int c) {               // 4 async b128 copies / thread
    const int buf = c & 1;
#pragma unroll
    for (int i = 0; i < 2; ++i) {
      __builtin_amdgcn_global_load_async_to_lds_b128(
          (v4i_t*)(xptr[i] + c * KCH), (lds_v4i_ptr)&XrStage[buf][t + 256 * i], 0, 0);
      __builtin_amdgcn_global_load_async_to_lds_b128(
          (v4i_t*)(wptr[i] + c * KCH), (lds_v4i_ptr)&WrStage[buf][t + 256 * i], 0, 0);
    }
  };
#endif

  auto stage_regs_to_tiles = [&](float4 xv, float4 wv, int i) {
    const int col = lrow[i] ^ (lkq[i] << 3);    // conflict-free scalar stores
    const int k0  = lkq[i] * 4;
    Xs[k0 + 0][col] = (xv.x < 0.0f) ? INF_F : xv.x;   // relu -> +inf
    Xs[k0 + 1][col] = (xv.y < 0.0f) ? INF_F : xv.y;
    Xs[k0 + 2][col] = (xv.z < 0.0f) ? INF_F : xv.z;
    Xs[k0 + 3][col] = (xv.w < 0.0f) ? INF_F : xv.w;
    Ws[k0 + 0][col] = wv.x;
    Ws[k0 + 1][col] = wv.y;
    Ws[k0 + 2][col] = wv.z;
    Ws[k0 + 3][col] = wv.w;
  };

  auto compute_chunk = [&]() {
    // Per k: 2x ds_load_b128 feed 16 v_add_f32 + 16 v_min_num_f32; adjacent
    // (i,j) updates are independent -> scheduler packs v_dual_add/v_dual_min.
#pragma unroll
    for (int k = 0; k < KCH; ++k) {
      const int swz = (k >> 2) << 3;
      const float4 xv = *(const float4*)&Xs[k][(ty * 4) ^ swz];
      const float4 wv = *(const float4*)&Ws[k][(tx * 4) ^ swz];
      const float xa[4] = {xv.x, xv.y, xv.z, xv.w};
      const float wa[4] = {wv.x, wv.y, wv.z, wv.w};
#pragma unroll
      for (int i = 0; i < 4; ++i)
#pragma unroll
        for (int j = 0; j < 4; ++j)
          acc[i][j] = fminf(acc[i][j], xa[i] + wa[j]);
    }
  };

  // ---- main pipeline -------------------------------------------------------
#if USE_ASYNC_LDS
  issue_chunk(0);
  for (int c = 0; c < nch - 1; ++c) {
    issue_chunk(c + 1);                         // copy next chunk (other buf)
    __builtin_amdgcn_s_wait_asynccnt(4);        // in-order: chunk c landed
    __syncthreads();                            // ...for every wave; prev compute done
    const int buf = c & 1;
#pragma unroll
    for (int i = 0; i < 2; ++i)
      stage_regs_to_tiles(XrStage[buf][t + 256 * i], WrStage[buf][t + 256 * i], i);
    __syncthreads();
    compute_chunk();
  }
  __builtin_amdgcn_s_wait_asynccnt(0);
  __syncthreads();
  {
    const int buf = (nch - 1) & 1;
#pragma unroll
    for (int i = 0; i < 2; ++i)
      stage_regs_to_tiles(XrStage[buf][t + 256 * i], WrStage[buf][t + 256 * i], i);
  }
  __syncthreads();
  compute_chunk();
#else
  // Fallback: register-staged software pipeline.
  float4 rx[2], rw[2];
#pragma unroll
  for (int i = 0; i < 2; ++i) {
    rx[i] = *(const float4*)(xptr[i]);
    rw[i] = *(const float4*)(wptr[i]);
  }
  for (int c = 0; c < nch; ++c) {
#pragma unroll
    for (int i = 0; i < 2; ++i)
      stage_regs_to_tiles(rx[i], rw[i], i);
    __syncthreads();
    if (c + 1 < nch) {
#pragma unroll
      for (int i = 0; i < 2; ++i) {
        rx[i] = *(const float4*)(xptr[i] + (c + 1) * KCH);
        rw[i] = *(const float4*)(wptr[i] + (c + 1) * KCH);
      }
    }
    compute_chunk();
    __syncthreads();
  }
#endif

  // ---- combine the two K halves with fp-min atomics ------------------------
  const int orow = b0 + ty * 4;
  const int ocol = o0 + tx * 4;
#pragma unroll
  for (int i = 0; i < 4; ++i)
#pragma unroll
    for (int j = 0; j < 4; ++j)
      atomic_fmin(out + (size_t)(orow + i) * O + (ocol + j), acc[i][j]);
}

extern "C" void kernel_launch(void* const* d_in, const int* in_sizes, int n_in,
                              void* d_out, int out_size, void* d_ws, size_t ws_size,
                              hipStream_t stream) {
  (void)n_in; (void)d_ws; (void)ws_size;
  const float* X = (const float*)d_in[0];        // [B, 512]
  const float* W = (const float*)d_in[1];        // [O, 512]
  float* out = (float*)d_out;                    // [B, O]
  const int B = in_sizes[0] / IN_F;              // 1024
  const int O = in_sizes[1] / IN_F;              // 512

  // out := +inf (atomic-min accumulators)
  const int n4 = out_size / 4;
  init_inf_kernel<<<(n4 + 255) / 256, 256, 0, stream>>>(out, n4);

  dim3 grid(O / TILE, B / TILE, SPLITK);         // (8, 16, 2) = 256 blocks
  TropicalMinReLu_kernel<<<grid, 256, 0, stream>>>(X, W, out, B, O);
}